// GNNModel_26809185861845
// MI455X (gfx1250) — compile-verified
//
#include <hip/hip_runtime.h>
#include <hip/hip_bf16.h>

// ---------------------------------------------------------------------------
// GCN (2x GraphConv, norm='both') + edge link predictor, fp32 end-to-end.
// GEMMs use CDNA5 V_WMMA_F32_16X16X4_F32 (native fp32 tensor op, no precision
// loss vs the reference). Edge gather/scatter is wave-per-edge with float4
// per lane; the 51MB feature matrix lives in the 192MB L2 so the random
// gathers and global_atomic_add_f32 scatters are L2-resident.
// ---------------------------------------------------------------------------

typedef __attribute__((ext_vector_type(2))) float v2f;
typedef __attribute__((ext_vector_type(8))) float v8f;

#define FEATS 128

// ---------------------------------------------------------------------------
// Degree accumulation: deg_out[src[e]] += 1, deg_in[dst[e]] += 1
// ---------------------------------------------------------------------------
__global__ void deg_kernel(const int* __restrict__ src,
                           const int* __restrict__ dst,
                           float* __restrict__ deg_out,
                           float* __restrict__ deg_in,
                           int n_edges) {
    int e = blockIdx.x * blockDim.x + threadIdx.x;
    if (e >= n_edges) return;
    atomicAdd(&deg_out[src[e]], 1.0f);
    atomicAdd(&deg_in[dst[e]], 1.0f);
}

// deg -> rsqrt(max(deg,1)) in place, for both arrays at once
__global__ void norm_kernel(float* __restrict__ deg_out,
                            float* __restrict__ deg_in,
                            int n_nodes) {
    int i = blockIdx.x * blockDim.x + threadIdx.x;
    if (i >= n_nodes) return;
    deg_out[i] = __frsqrt_rn(fmaxf(deg_out[i], 1.0f));
    deg_in[i]  = __frsqrt_rn(fmaxf(deg_in[i], 1.0f));
}

// ---------------------------------------------------------------------------
// C[M,128] = A[M,128] @ W[128,128] using V_WMMA_F32_16X16X4_F32.
// Block = 256 threads = 8 wave32s. Each block computes a 16-row strip,
// each wave one 16x16 tile (wave w -> columns [16w,16w+16)).
// W (64KB) is staged in LDS once per block (WGP has 320KB).
//
// Fragment layouts (ISA 7.12.2, 32-bit data):
//   A 16x4 : lanes 0-15 M=lane K={0,1} in v[0],v[1]; lanes 16-31 K={2,3}
//   B 4x16 : lanes 0-15 N=lane K={0,1};              lanes 16-31 K={2,3}
//   C 16x16: vgpr v, lane l: M = v + 8*(l>=16), N = l%16
// ---------------------------------------------------------------------------
__global__ void __launch_bounds__(256)
gemm128_wmma(const float* __restrict__ A,
             const float* __restrict__ W,
             float* __restrict__ C,
             int M) {
    __shared__ float sW[FEATS * FEATS];   // 64 KB

    // cooperative 128-bit copy of W into LDS
    const float4* Wv = (const float4*)W;
    float4* sWv = (float4*)sW;
    for (int i = threadIdx.x; i < FEATS * FEATS / 4; i += blockDim.x)
        sWv[i] = Wv[i];
    __syncthreads();

    const int row0 = blockIdx.x * 16;
    const int wave = threadIdx.x >> 5;        // 0..7 -> column tile
    const int lane = threadIdx.x & 31;
    const int col0 = wave * 16;
    const int half = lane >> 4;               // 0: K={0,1}, 1: K={2,3}
    const int l16  = lane & 15;

    int arow = row0 + l16;
    if (arow >= M) arow = M - 1;              // clamp (stores are guarded)
    const float* __restrict__ ap = A + (size_t)arow * FEATS;

    v8f acc = {};
    #pragma unroll 4
    for (int kk = 0; kk < FEATS; kk += 4) {
        const int k0 = kk + 2 * half;
        v2f a, b;
        a.x = ap[k0];
        a.y = ap[k0 + 1];
        b.x = sW[k0 * FEATS + col0 + l16];
        b.y = sW[(k0 + 1) * FEATS + col0 + l16];
        acc = __builtin_amdgcn_wmma_f32_16x16x4_f32(
            /*neg_a=*/false, a, /*neg_b=*/false, b,
            /*c_mod=*/(short)0, acc, /*reuse_a=*/false, /*reuse_b=*/false);
    }

    // Epilogue: the bounds check is block-uniform (full 16-row strip in
    // range or not), so take a branch-free fast path in the common case.
    float* __restrict__ cp = C + ((size_t)(row0 + 8 * half)) * FEATS + col0 + l16;
    if (row0 + 16 <= M) {
        #pragma unroll
        for (int v = 0; v < 8; ++v)
            cp[(size_t)v * FEATS] = acc[v];
    } else {
        #pragma unroll
        for (int v = 0; v < 8; ++v) {
            if (row0 + v + 8 * half < M) cp[(size_t)v * FEATS] = acc[v];
        }
    }
}

// ---------------------------------------------------------------------------
// Edge scatter: agg[dst[e]] += h[src[e]] * norm_src[src[e]]
// One wave32 per edge, one float4 (4 feats) per lane -> 128 feats.
// ---------------------------------------------------------------------------
__global__ void scatter_kernel(const float* __restrict__ h,
                               const float* __restrict__ norm_src,
                               const int* __restrict__ src,
                               const int* __restrict__ dst,
                               float* __restrict__ agg,
                               int n_edges) {
    const int gid  = blockIdx.x * blockDim.x + threadIdx.x;
    const int edge = gid >> 5;
    const int lane = gid & 31;
    if (edge >= n_edges) return;

    const int s = src[edge];
    const int d = dst[edge];
    const float ns = norm_src[s];

    const float4 v = ((const float4*)(h + (size_t)s * FEATS))[lane];
    float* out = agg + (size_t)d * FEATS + lane * 4;
    atomicAdd(out + 0, v.x * ns);
    atomicAdd(out + 1, v.y * ns);
    atomicAdd(out + 2, v.z * ns);
    atomicAdd(out + 3, v.w * ns);
}

// ---------------------------------------------------------------------------
// out[i] = act(agg[i] * norm_dst[node] + bias[feat]); act = relu or identity
// ---------------------------------------------------------------------------
template <bool RELU>
__global__ void bias_act_kernel(const float* __restrict__ agg,
                                const float* __restrict__ norm_dst,
                                const float* __restrict__ bias,
                                float* __restrict__ out,
                                int total) {
    const int i = blockIdx.x * blockDim.x + threadIdx.x;
    if (i >= total) return;
    const int node = i >> 7;       // /128
    const int f    = i & (FEATS - 1);
    float v = fmaf(agg[i], norm_dst[node], bias[f]);
    if (RELU) v = fmaxf(v, 0.0f);
    out[i] = v;
}

// ---------------------------------------------------------------------------
// Link predictor: score[e] = sigmoid( dot(h[src],Wp[0:128])
//                                   + dot(h[dst],Wp[128:256]) + bp )
// One wave32 per edge; shuffle tree reduction (wave32-native).
// ---------------------------------------------------------------------------
__global__ void predict_kernel(const float* __restrict__ h,
                               const float* __restrict__ Wp,
                               const float* __restrict__ bp,
                               const int* __restrict__ src,
                               const int* __restrict__ dst,
                               float* __restrict__ out,
                               int n_edges) {
    const int gid  = blockIdx.x * blockDim.x + threadIdx.x;
    const int edge = gid >> 5;
    const int lane = gid & 31;
    if (edge >= n_edges) return;

    const int s = src[edge];
    const int d = dst[edge];

    const float4 hs = ((const float4*)(h + (size_t)s * FEATS))[lane];
    const float4 hd = ((const float4*)(h + (size_t)d * FEATS))[lane];
    const float4 ws = ((const float4*)Wp)[lane];
    const float4 wd = ((const float4*)(Wp + FEATS))[lane];

    float acc = hs.x * ws.x + hs.y * ws.y + hs.z * ws.z + hs.w * ws.w
              + hd.x * wd.x + hd.y * wd.y + hd.z * wd.z + hd.w * wd.w;

    #pragma unroll
    for (int off = 16; off > 0; off >>= 1)
        acc += __shfl_down(acc, off, 32);

    if (lane == 0) {
        const float z = acc + bp[0];
        out[edge] = 1.0f / (1.0f + __expf(-z));
    }
}

// ---------------------------------------------------------------------------
// Launch
// ---------------------------------------------------------------------------
extern "C" void kernel_launch(void* const* d_in, const int* in_sizes, int n_in,
                              void* d_out, int out_size, void* d_ws, size_t ws_size,
                              hipStream_t stream) {
    const float* x  = (const float*)d_in[0];
    const float* W1 = (const float*)d_in[1];
    const float* b1 = (const float*)d_in[2];
    const float* W2 = (const float*)d_in[3];
    const float* b2 = (const float*)d_in[4];
    const float* Wp = (const float*)d_in[5];
    const float* bp = (const float*)d_in[6];
    const int* src  = (const int*)d_in[7];
    const int* dst  = (const int*)d_in[8];
    float* scores   = (float*)d_out;

    const int n_nodes = in_sizes[0] / FEATS;
    const int n_edges = in_sizes[7];
    const size_t feat_bytes = (size_t)n_nodes * FEATS * sizeof(float);
    const size_t norm_bytes = ((size_t)n_nodes * sizeof(float) + 255) & ~(size_t)255;

    char* ws = (char*)d_ws;
    float* norm_src = (float*)(ws);                      // deg_out -> rsqrt
    float* norm_dst = (float*)(ws + norm_bytes);         // deg_in  -> rsqrt
    float* bufA     = (float*)(ws + 2 * norm_bytes);
    float* bufB     = (float*)(ws + 2 * norm_bytes + feat_bytes);

    const int TPB = 256;
    const int eb   = (n_edges + TPB - 1) / TPB;            // thread-per-edge
    const int nb   = (n_nodes + TPB - 1) / TPB;            // thread-per-node
    const int ewb  = (n_edges * 32 + TPB - 1) / TPB;       // wave-per-edge
    const int fb   = (n_nodes * FEATS + TPB - 1) / TPB;    // thread-per-feat
    const int gemb = (n_nodes + 15) / 16;                  // 16-row strips

    // degrees + norms
    hipMemsetAsync(norm_src, 0, n_nodes * sizeof(float), stream);
    hipMemsetAsync(norm_dst, 0, n_nodes * sizeof(float), stream);
    deg_kernel<<<eb, TPB, 0, stream>>>(src, dst, norm_src, norm_dst, n_edges);
    norm_kernel<<<nb, TPB, 0, stream>>>(norm_src, norm_dst, n_nodes);

    // ---- layer 1: h1 = relu(normdst * scatter(normsrc * (x@W1)) + b1) ----
    gemm128_wmma<<<gemb, TPB, 0, stream>>>(x, W1, bufA, n_nodes);
    hipMemsetAsync(bufB, 0, feat_bytes, stream);
    scatter_kernel<<<ewb, TPB, 0, stream>>>(bufA, norm_src, src, dst, bufB, n_edges);
    bias_act_kernel<true><<<fb, TPB, 0, stream>>>(bufB, norm_dst, b1, bufA,
                                                  n_nodes * FEATS);

    // ---- layer 2: h2 = normdst * scatter(normsrc * (h1@W2)) + b2 ----
    gemm128_wmma<<<gemb, TPB, 0, stream>>>(bufA, W2, bufB, n_nodes);
    hipMemsetAsync(bufA, 0, feat_bytes, stream);
    scatter_kernel<<<ewb, TPB, 0, stream>>>(bufB, norm_src, src, dst, bufA, n_edges);
    bias_act_kernel<false><<<fb, TPB, 0, stream>>>(bufA, norm_dst, b2, bufB,
                                                   n_nodes * FEATS);

    // ---- link predictor ----
    predict_kernel<<<ewb, TPB, 0, stream>>>(bufB, Wp, bp, src, dst, scores, n_edges);
}